// StructuredElmanCell_55353538510898
// MI455X (gfx1250) — compile-verified
//
#include <hip/hip_runtime.h>
#include <math.h>

// Problem constants (from reference):
#define T_  1024
#define BS_ 4
#define NH_ 16
#define DS_ 32
#define HD_ 128
#define R_  8
#define DI_ (NH_*HD_)   // 2048
#define PF_ 16          // L2 prefetch distance in timesteps

// Per-timestep strides (in floats) -- compile-time constants.
#define SB_ (BS_*NH_*DS_*R_)   // B_proj   : 16384
#define SX_ (BS_*NH_*HD_*R_)   // X_proj   : 65536
#define SA_ (BS_*NH_)          // alpha_raw: 64
#define SZ_ (BS_*DI_)          // z        : 8192
#define SO_ (BS_*DI_)          // out      : 8192

typedef __attribute__((ext_vector_type(2))) float v2f;
typedef __attribute__((ext_vector_type(8))) float v8f;

#if __has_builtin(__builtin_amdgcn_tanhf)
__device__ __forceinline__ float fast_tanh(float x) { return __builtin_amdgcn_tanhf(x); }
#else
__device__ __forceinline__ float fast_tanh(float x) { return tanhf(x); }
#endif
__device__ __forceinline__ float fast_exp(float x) { return __expf(x); }
__device__ __forceinline__ float fast_rcp(float x) { return __builtin_amdgcn_rcpf(x); }
// silu(x) = x*sigmoid(x) = h + h*tanh(h), h = x/2 : one v_tanh_f32 + one FMA.
__device__ __forceinline__ float silu_f(float x) {
    const float h = 0.5f * x;
    const float t = fast_tanh(h);
    return __builtin_fmaf(h, t, h);
}

// Swap lane halves (0-15 <-> 16-31) with a single VALU v_permlanex16_b32
// (identity nibble selectors) instead of an LDS ds_bpermute round-trip.
__device__ __forceinline__ float half_swap(float v) {
#if __has_builtin(__builtin_amdgcn_permlanex16)
    const int i = __builtin_bit_cast(int, v);
    const int r = __builtin_amdgcn_permlanex16(i, i, 0x76543210, 0xfedcba98,
                                               false, false);
    return __builtin_bit_cast(float, r);
#else
    return __shfl_xor(v, 16, 32);
#endif
}

// Per-step register staging for one wave (one 32x16 slab of H): 14 VGPRs.
struct StepRegs {
    v2f  A[2][2];   // [m_tile][k_chunk] : B_t fragments
    v2f  Bf[2];     // [k_chunk]         : X_t fragments (single N-tile)
    float araw;
    float zv;
};

// One single-wave workgroup per (b, h, n_tile): 64 chains x 8 N-tiles = 512
// independent waves; the scan is column-separable, so the only cross-WG
// redundancy is re-reading the tiny shared B_t (hits L2).
__global__ __launch_bounds__(32)
void elman_scan_kernel(const float* __restrict__ B_proj,    // [T,BS,NH,DS,R]
                       const float* __restrict__ X_proj,    // [T,BS,NH,HD,R]
                       const float* __restrict__ alpha_raw, // [T,BS,NH]
                       const float* __restrict__ alpha_bias,// [NH]
                       const float* __restrict__ z,         // [T,BS,DI]
                       const float* __restrict__ H0,        // [BS,NH,DS,HD]
                       float* __restrict__ out,             // [T,BS,DI]
                       float* __restrict__ Hfin)            // [BS,NH,DS,HD]
{
    const int bid   = blockIdx.x;        // 0..511
    const int chain = bid >> 3;          // b*NH + h
    const int ntile = bid & 7;
    const int b     = chain / NH_;
    const int h     = chain % NH_;
    const int n0    = ntile * 16;
    const int lane  = threadIdx.x & 31;
    const int l16   = lane & 15;
    const int rhi   = (lane >> 4) << 1;  // A/B frag: lanes 16-31 hold K+2
    const int mhi   = (lane >> 4) << 3;  // C/D frag: lanes 16-31 hold M+8

    const float ab = alpha_bias[h];

    // H state in WMMA C/D layout: Hreg[mt][i] = H(M = mt*16+i+mhi, N = n0+l16).
    v8f Hreg[2];
    #pragma unroll
    for (int mt = 0; mt < 2; ++mt)
        #pragma unroll
        for (int i = 0; i < 8; ++i) {
            const int m = mt*16 + i + mhi;
            Hreg[mt][i] = H0[(((size_t)chain)*DS_ + m)*HD_ + n0 + l16];
        }

    // Per-lane base pointers; advanced by 4 timesteps per loop iteration so all
    // in-loop accesses use compile-time stage offsets (fit 24-bit IOFFSET).
    const float* pB = B_proj    + (size_t)chain*(DS_*R_) + l16*R_ + rhi;
    const float* pX = X_proj    + (size_t)chain*(HD_*R_) + (n0 + l16)*R_ + rhi;
    const float* pA = alpha_raw + chain;
    const float* pZ = z         + (size_t)b*DI_ + h*HD_ + n0 + l16;
    float*       pO = out       + (size_t)b*DI_ + h*HD_ + n0 + l16;

    // Issue (unwaited) fragment loads for pipeline stage k (k compile-time).
    auto issue_k = [&](int k, StepRegs& s) {
        #pragma unroll
        for (int mt = 0; mt < 2; ++mt)
            #pragma unroll
            for (int kc = 0; kc < 2; ++kc)
                s.A[mt][kc] = *(const v2f*)(pB + k*SB_ + mt*(16*R_) + kc*4);
        #pragma unroll
        for (int kc = 0; kc < 2; ++kc)
            s.Bf[kc] = *(const v2f*)(pX + k*SX_ + kc*4);
        s.araw = pA[k*SA_];
        s.zv   = pZ[k*SZ_];
    };

    auto compute_k = [&](int k, const StepRegs& s) {
        // alpha = sigmoid(-softplus(raw + bias)) == 1/(2 + exp(raw + bias))
        // exactly (exp(softplus(x)) = 1 + e^x). Uniform -> v_s_exp + v_s_rcp.
        const float xr = s.araw + ab;
        const float a  = fast_rcp(2.f + fast_exp(xr));

        // c = a*H + B_t @ X_t^T : seed C, two chained K=4 WMMAs per M-tile.
        v8f c[2];
        #pragma unroll
        for (int mt = 0; mt < 2; ++mt)
            #pragma unroll
            for (int i = 0; i < 8; ++i) c[mt][i] = a * Hreg[mt][i];
        #pragma unroll
        for (int mt = 0; mt < 2; ++mt) {
            c[mt] = __builtin_amdgcn_wmma_f32_16x16x4_f32(
                        false, s.A[mt][0], false, s.Bf[0], (short)0, c[mt], false, false);
            c[mt] = __builtin_amdgcn_wmma_f32_16x16x4_f32(
                        false, s.A[mt][1], false, s.Bf[1], (short)0, c[mt], false, false);
        }

        // H = silu(c); y = sum over rows: 16 in-register (log-depth tree) plus
        // the other lane half via one v_permlanex16.
        v8f sv;
        #pragma unroll
        for (int i = 0; i < 8; ++i) {
            const float v = silu_f(c[0][i]);
            Hreg[0][i] = v;  sv[i] = v;
        }
        #pragma unroll
        for (int i = 0; i < 8; ++i) {
            const float v = silu_f(c[1][i]);
            Hreg[1][i] = v;  sv[i] += v;
        }
        const float a0 = sv[0] + sv[4], a1 = sv[1] + sv[5];
        const float a2 = sv[2] + sv[6], a3 = sv[3] + sv[7];
        float acc = (a0 + a2) + (a1 + a3);
        acc += half_swap(acc);

        // Both lane halves now hold identical y for column n0+l16: store from
        // all 32 lanes (duplicate same-data stores; branch-free, EXEC stays 1s).
        const float y  = acc;
        const float zy = s.zv + y;
        pO[k*SO_] = y * silu_f(zy);

        // Streaming speculative L2 prefetch (OOB near the end is dropped by HW).
        __builtin_prefetch(pX + (PF_ + k)*SX_, 0, 0);
        __builtin_prefetch(pB + (PF_ + k)*SB_, 0, 0);
    };

    // 4-deep software pipeline, unrolled x4; last group peeled so the steady
    // state has no index clamping (pure constant offsets + pointer bumps).
    StepRegs r0, r1, r2, r3;
    issue_k(0, r0); issue_k(1, r1); issue_k(2, r2); issue_k(3, r3);

    for (int it = 0; it < T_/4 - 1; ++it) {
        compute_k(0, r0); issue_k(4, r0);
        compute_k(1, r1); issue_k(5, r1);
        compute_k(2, r2); issue_k(6, r2);
        compute_k(3, r3); issue_k(7, r3);
        pB += 4*SB_; pX += 4*SX_; pA += 4*SA_; pZ += 4*SZ_; pO += 4*SO_;
    }
    // Tail: last 4 timesteps, nothing left to issue.
    compute_k(0, r0); compute_k(1, r1); compute_k(2, r2); compute_k(3, r3);

    // Store H_final from fragments.
    #pragma unroll
    for (int mt = 0; mt < 2; ++mt)
        #pragma unroll
        for (int i = 0; i < 8; ++i) {
            const int m = mt*16 + i + mhi;
            Hfin[(((size_t)chain)*DS_ + m)*HD_ + n0 + l16] = Hreg[mt][i];
        }
}

extern "C" void kernel_launch(void* const* d_in, const int* in_sizes, int n_in,
                              void* d_out, int out_size, void* d_ws, size_t ws_size,
                              hipStream_t stream) {
    (void)in_sizes; (void)n_in; (void)out_size; (void)d_ws; (void)ws_size;
    const float* B_proj     = (const float*)d_in[0];
    const float* X_proj     = (const float*)d_in[1];
    const float* alpha_raw  = (const float*)d_in[2];
    const float* alpha_bias = (const float*)d_in[3];
    const float* z          = (const float*)d_in[4];
    const float* H0         = (const float*)d_in[5];

    float* out  = (float*)d_out;                    // [T,BS,DI] first
    float* Hfin = out + (size_t)T_ * BS_ * DI_;     // then [BS,NH,DS,HD]

    elman_scan_kernel<<<dim3(BS_ * NH_ * 8), dim3(32), 0, stream>>>(
        B_proj, X_proj, alpha_raw, alpha_bias, z, H0, out, Hfin);
}